// NeuralODE_13082470383967
// MI455X (gfx1250) — compile-verified
//
#include <hip/hip_runtime.h>
#include <hip/hip_bf16.h>

typedef float v2f __attribute__((ext_vector_type(2)));
typedef float v4f __attribute__((ext_vector_type(4)));
typedef float v8f __attribute__((ext_vector_type(8)));

#define B_N 4096
#define D_N 64
#define H_N 32
#define TRAJ_LEN 11
#define MAX_INNER 64
#define N_INTERVALS 10
#define TILE_M 16
#define YS 68   // padded row stride (floats) for 64-wide buffers -> conflict-free banks, 16B-aligned rows
#define HS 36   // padded row stride (floats) for 32-wide buffers

#define RTOL_F 1e-3f
#define ATOL_F 1e-6f
#define DT0_F  1e-3f

// Tsit5 tableau (f32)
#define A21f 0.161f
#define A31f (-0.008480655492356989f)
#define A32f 0.335480655492357f
#define A41f 2.8971530571054935f
#define A42f (-6.359448489975075f)
#define A43f 4.3622954328695815f
#define A51f 5.325864828439257f
#define A52f (-11.748883564062828f)
#define A53f 7.4955393428898365f
#define A54f (-0.09249506636175525f)
#define A61f 5.86145544294642f
#define A62f (-12.92096931784711f)
#define A63f 8.159367898576159f
#define A64f (-0.071584973281401f)
#define A65f (-0.028269050394068383f)
#define B1f 0.09646076681806523f
#define B2f 0.01f
#define B3f 0.4798896504144996f
#define B4f 1.379008574103742f
#define B5f (-3.290069515436081f)
#define B6f 2.324710524099774f
#define E1f (-0.00178001105222577714f)
#define E2f (-0.0008164344596567469f)
#define E3f 0.007880878010261995f
#define E4f (-0.1447110071732629f)
#define E5f 0.5823571654525552f
#define E6f (-0.45808210592918697f)
#define E7f 0.015151515151515152f

__device__ __forceinline__ float softplus_f(float x) {
    // jax.nn.softplus: max(x,0) + log1p(exp(-|x|))
    return fmaxf(x, 0.0f) + log1pf(expf(-fabsf(x)));
}

__device__ __forceinline__ v8f wmma4(v2f a, v2f b, v8f c) {
    // D = A(16x4 f32) * B(4x16 f32) + C(16x16 f32); full f32 matrix-pipe op
    return __builtin_amdgcn_wmma_f32_16x16x4_f32(false, a, false, b, (short)0, c,
                                                 false, false);
}

__device__ __forceinline__ v4f ld4(const float* p) { return *(const v4f*)p; }
__device__ __forceinline__ void st4(float* p, v4f v) { *(v4f*)p = v; }
__device__ __forceinline__ v4f vabs4(v4f x) {
    v4f r;
#pragma unroll
    for (int i = 0; i < 4; ++i) r[i] = fabsf(x[i]);
    return r;
}
__device__ __forceinline__ v4f vmax4(v4f a, v4f b) {
    v4f r;
#pragma unroll
    for (int i = 0; i < 4; ++i) r[i] = fmaxf(a[i], b[i]);
    return r;
}

struct WFrags {
    v2f w1[16][2];  // W1^T [64x32] as B-frags: 16 K-chunks x 2 N-tiles
    v2f w2[8][2];   // W2^T [32x32]
    v2f w3[8][4];   // W3^T [32x64]
    float b1[2], b2[2], b3[4];
};

// One MLP evaluation for a 16-sample tile: yin (LDS, [16][YS]) -> out (C-layout regs, 16x64)
__device__ __forceinline__ void mlp_stage(const WFrags& F,
                                          const float* __restrict__ yin,
                                          float* __restrict__ hb1,
                                          float* __restrict__ hb2,
                                          v8f out[4], int nl, int koff, int mbase) {
    const v8f z = {0.f,0.f,0.f,0.f,0.f,0.f,0.f,0.f};
    v8f h[2] = {z, z};
#pragma unroll
    for (int c = 0; c < 16; ++c) {
        v2f a = *(const v2f*)(yin + nl * YS + 4 * c + koff);
        h[0] = wmma4(a, F.w1[c][0], h[0]);
        h[1] = wmma4(a, F.w1[c][1], h[1]);
    }
#pragma unroll
    for (int t = 0; t < 2; ++t) {
#pragma unroll
        for (int v = 0; v < 8; ++v)
            hb1[(v + mbase) * HS + t * 16 + nl] = softplus_f(h[t][v] + F.b1[t]);
    }
    __syncthreads();
    v8f g[2] = {z, z};
#pragma unroll
    for (int c = 0; c < 8; ++c) {
        v2f a = *(const v2f*)(hb1 + nl * HS + 4 * c + koff);
        g[0] = wmma4(a, F.w2[c][0], g[0]);
        g[1] = wmma4(a, F.w2[c][1], g[1]);
    }
#pragma unroll
    for (int t = 0; t < 2; ++t) {
#pragma unroll
        for (int v = 0; v < 8; ++v)
            hb2[(v + mbase) * HS + t * 16 + nl] = softplus_f(g[t][v] + F.b2[t]);
    }
    __syncthreads();
    out[0] = z; out[1] = z; out[2] = z; out[3] = z;
#pragma unroll
    for (int c = 0; c < 8; ++c) {
        v2f a = *(const v2f*)(hb2 + nl * HS + 4 * c + koff);
#pragma unroll
        for (int n = 0; n < 4; ++n)
            out[n] = wmma4(a, F.w3[c][n], out[n]);
    }
#pragma unroll
    for (int n = 0; n < 4; ++n)
#pragma unroll
        for (int v = 0; v < 8; ++v)
            out[n][v] += F.b3[n];
    __syncthreads();  // safe reuse of hb buffers / yin by caller
}

__device__ __forceinline__ void storeC(const v8f k[4], float* __restrict__ buf,
                                       int nl, int mbase) {
#pragma unroll
    for (int t = 0; t < 4; ++t)
#pragma unroll
        for (int v = 0; v < 8; ++v)
            buf[(v + mbase) * YS + t * 16 + nl] = k[t][v];
}

// vectorized (b128 LDS) elementwise pass over this thread's 32 state elements
#define ROWLOOP4(expr)                                             \
    {                                                              \
        const int r_ = nl;                                         \
        const float hm = Sh[r_];                                   \
        (void)hm;                                                  \
        _Pragma("unroll")                                          \
        for (int j = 0; j < 8; ++j) {                              \
            const int o = r_ * YS + ch * 32 + 4 * j;               \
            expr;                                                  \
        }                                                          \
    }

__global__ void __launch_bounds__(32)
node_tsit5_kernel(const float* __restrict__ x0s,
                  const float* __restrict__ W1, const float* __restrict__ b1,
                  const float* __restrict__ W2, const float* __restrict__ b2,
                  const float* __restrict__ W3, const float* __restrict__ b3,
                  const int* __restrict__ Tp,
                  float* __restrict__ traj, int* __restrict__ nsum, int has_nsum) {
    __shared__ float Y[TILE_M * YS];
    __shared__ float YT[TILE_M * YS];
    __shared__ float K1[TILE_M * YS];
    __shared__ float K2[TILE_M * YS];
    __shared__ float K3[TILE_M * YS];
    __shared__ float K4[TILE_M * YS];
    __shared__ float K5[TILE_M * YS];
    __shared__ float K6[TILE_M * YS];
    __shared__ float HB1[TILE_M * HS];
    __shared__ float HB2[TILE_M * HS];
    __shared__ float St[TILE_M], Sdt[TILE_M], Sh[TILE_M];
    __shared__ int Sn[TILE_M], Sdone[TILE_M], Sok[TILE_M];

    const int lane = threadIdx.x;
    const int nl = lane & 15;
    const int koff = (lane >> 4) << 1;   // 0 for lanes 0-15, 2 for lanes 16-31
    const int mbase = (lane >> 4) << 3;  // C-layout row offset
    const int ch = lane >> 4;            // column half for elementwise loops
    const int base = blockIdx.x * TILE_M;

    // ---- Load weights once into B-fragment registers ----
    WFrags F;
#pragma unroll
    for (int c = 0; c < 16; ++c)
#pragma unroll
        for (int n = 0; n < 2; ++n)
            F.w1[c][n] = *(const v2f*)(W1 + (n * 16 + nl) * D_N + 4 * c + koff);
#pragma unroll
    for (int c = 0; c < 8; ++c)
#pragma unroll
        for (int n = 0; n < 2; ++n)
            F.w2[c][n] = *(const v2f*)(W2 + (n * 16 + nl) * H_N + 4 * c + koff);
#pragma unroll
    for (int c = 0; c < 8; ++c)
#pragma unroll
        for (int n = 0; n < 4; ++n)
            F.w3[c][n] = *(const v2f*)(W3 + (n * 16 + nl) * H_N + 4 * c + koff);
    F.b1[0] = b1[nl]; F.b1[1] = b1[16 + nl];
    F.b2[0] = b2[nl]; F.b2[1] = b2[16 + nl];
#pragma unroll
    for (int n = 0; n < 4; ++n) F.b3[n] = b3[n * 16 + nl];

    const float Ttot = (float)Tp[0];

    // ---- Init state + traj[:,0,:] (vectorized b128) ----
    {
        const int r = nl;
#pragma unroll
        for (int j = 0; j < 8; ++j) {
            const int d = ch * 32 + 4 * j;
            const v4f v = ld4(x0s + (size_t)(base + r) * D_N + d);
            st4(&Y[r * YS + d], v);
            st4(traj + (size_t)(base + r) * TRAJ_LEN * D_N + d, v);
        }
    }
    if (lane < TILE_M) { St[lane] = 0.0f; Sdt[lane] = DT0_F; Sn[lane] = 0; }
    __syncthreads();

    v8f k[4];

#pragma unroll 1
    for (int iv = 0; iv < N_INTERVALS; ++iv) {
        const float t_target = Ttot * (float)(iv + 1) * 0.1f;
#pragma unroll 1
        for (int step = 0; step < MAX_INNER; ++step) {
            // per-sample step-size control
            if (lane < TILE_M) {
                const float rem = t_target - St[lane];
                Sdone[lane] = (rem <= 1e-12f) ? 1 : 0;
                Sh[lane] = fminf(Sdt[lane], fmaxf(rem, 0.0f));
            }
            __syncthreads();

            // k1 = f(y)
            mlp_stage(F, Y, HB1, HB2, k, nl, koff, mbase);
            storeC(k, K1, nl, mbase);
            __syncthreads();
            ROWLOOP4(st4(&YT[o], ld4(&Y[o]) + hm * (A21f * ld4(&K1[o]))));
            __syncthreads();
            // k2
            mlp_stage(F, YT, HB1, HB2, k, nl, koff, mbase);
            storeC(k, K2, nl, mbase);
            __syncthreads();
            ROWLOOP4(st4(&YT[o], ld4(&Y[o]) + hm * (A31f * ld4(&K1[o]) +
                                                    A32f * ld4(&K2[o]))));
            __syncthreads();
            // k3
            mlp_stage(F, YT, HB1, HB2, k, nl, koff, mbase);
            storeC(k, K3, nl, mbase);
            __syncthreads();
            ROWLOOP4(st4(&YT[o], ld4(&Y[o]) + hm * (A41f * ld4(&K1[o]) +
                                                    A42f * ld4(&K2[o]) +
                                                    A43f * ld4(&K3[o]))));
            __syncthreads();
            // k4
            mlp_stage(F, YT, HB1, HB2, k, nl, koff, mbase);
            storeC(k, K4, nl, mbase);
            __syncthreads();
            ROWLOOP4(st4(&YT[o], ld4(&Y[o]) + hm * (A51f * ld4(&K1[o]) +
                                                    A52f * ld4(&K2[o]) +
                                                    A53f * ld4(&K3[o]) +
                                                    A54f * ld4(&K4[o]))));
            __syncthreads();
            // k5
            mlp_stage(F, YT, HB1, HB2, k, nl, koff, mbase);
            storeC(k, K5, nl, mbase);
            __syncthreads();
            ROWLOOP4(st4(&YT[o], ld4(&Y[o]) + hm * (A61f * ld4(&K1[o]) +
                                                    A62f * ld4(&K2[o]) +
                                                    A63f * ld4(&K3[o]) +
                                                    A64f * ld4(&K4[o]) +
                                                    A65f * ld4(&K5[o]))));
            __syncthreads();
            // k6
            mlp_stage(F, YT, HB1, HB2, k, nl, koff, mbase);
            storeC(k, K6, nl, mbase);
            __syncthreads();
            // Fused pass: y5 -> YT  AND  E-partial (E1..E6 dot k1..k6) -> K1
            ROWLOOP4({
                const v4f k1 = ld4(&K1[o]);
                const v4f k2 = ld4(&K2[o]);
                const v4f k3 = ld4(&K3[o]);
                const v4f k4 = ld4(&K4[o]);
                const v4f k5 = ld4(&K5[o]);
                const v4f k6 = ld4(&K6[o]);
                st4(&YT[o], ld4(&Y[o]) + hm * (B1f * k1 + B2f * k2 + B3f * k3 +
                                               B4f * k4 + B5f * k5 + B6f * k6));
                st4(&K1[o], E1f * k1 + E2f * k2 + E3f * k3 +
                            E4f * k4 + E5f * k5 + E6f * k6);
            });
            __syncthreads();
            // k7 = f(y5); spill into K2 (dead)
            mlp_stage(F, YT, HB1, HB2, k, nl, koff, mbase);
            storeC(k, K2, nl, mbase);
            __syncthreads();

            // squared error ratios -> K1 (4-stream vectorized pass)
            ROWLOOP4({
                const v4f e = hm * (ld4(&K1[o]) + E7f * ld4(&K2[o]));
                const v4f sc =
                    ATOL_F + RTOL_F * vmax4(vabs4(ld4(&Y[o])), vabs4(ld4(&YT[o])));
                const v4f rr = e / sc;
                st4(&K1[o], rr * rr);
            });
            __syncthreads();

            // per-sample accept/step-size update (lanes 0-15), vectorized reduction
            if (lane < TILE_M) {
                v4f acc = {0.f, 0.f, 0.f, 0.f};
#pragma unroll
                for (int q = 0; q < 16; ++q) acc += ld4(&K1[lane * YS + 4 * q]);
                const float s = acc.x + acc.y + acc.z + acc.w;
                const float enorm = fmaxf(sqrtf(s * (1.0f / 64.0f)), 1e-10f);
                const int accept = (enorm <= 1.0f) ? 1 : 0;
                const float fac =
                    fminf(fmaxf(0.9f * powf(enorm, -0.2f), 0.1f), 5.0f);
                const int done = Sdone[lane];
                const float hm = Sh[lane];
                const int ok = (accept && !done) ? 1 : 0;
                Sok[lane] = ok;
                if (ok) St[lane] = St[lane] + hm;
                if (!done) {
                    Sdt[lane] = fmaxf(hm * fac, 1e-8f);
                    Sn[lane] += 1;
                }
            }
            __syncthreads();
            // y := y5 where accepted (vectorized)
            {
                const int r = nl;
                if (Sok[r]) {
#pragma unroll
                    for (int j = 0; j < 8; ++j) {
                        const int o = r * YS + ch * 32 + 4 * j;
                        st4(&Y[o], ld4(&YT[o]));
                    }
                }
            }
            __syncthreads();
        }
        // save interval endpoint (vectorized global stores)
        {
            const int r = nl;
#pragma unroll
            for (int j = 0; j < 8; ++j) {
                const int d = ch * 32 + 4 * j;
                st4(traj + (size_t)(base + r) * TRAJ_LEN * D_N +
                        (size_t)(iv + 1) * D_N + d,
                    ld4(&Y[r * YS + d]));
            }
        }
        __syncthreads();
    }

    if (has_nsum && lane < TILE_M) atomicAdd(nsum, Sn[lane]);
}

__global__ void zero_nsum_kernel(int* p) {
    if (threadIdx.x == 0 && blockIdx.x == 0) *p = 0;
}

extern "C" void kernel_launch(void* const* d_in, const int* in_sizes, int n_in,
                              void* d_out, int out_size, void* d_ws, size_t ws_size,
                              hipStream_t stream) {
    (void)in_sizes; (void)n_in; (void)d_ws; (void)ws_size;
    const float* x0s = (const float*)d_in[0];
    const float* W1  = (const float*)d_in[1];
    const float* b1  = (const float*)d_in[2];
    const float* W2  = (const float*)d_in[3];
    const float* b2  = (const float*)d_in[4];
    const float* W3  = (const float*)d_in[5];
    const float* b3  = (const float*)d_in[6];
    const int*   Tp  = (const int*)d_in[7];

    float* traj = (float*)d_out;
    const int traj_elems = B_N * TRAJ_LEN * D_N;
    const int has_nsum = (out_size > traj_elems) ? 1 : 0;
    int* nsum = (int*)(traj + traj_elems);

    if (has_nsum) zero_nsum_kernel<<<1, 32, 0, stream>>>(nsum);
    node_tsit5_kernel<<<B_N / TILE_M, 32, 0, stream>>>(
        x0s, W1, b1, W2, b2, W3, b3, Tp, traj, nsum, has_nsum);
}